// QuantumGateBlock_65481071405546
// MI455X (gfx1250) — compile-verified
//
#include <hip/hip_runtime.h>
#include <cstdint>

#define NQ 20
#define NSAMP 64

typedef float v2f __attribute__((ext_vector_type(2)));
typedef float v8f __attribute__((ext_vector_type(8)));

__device__ __forceinline__ float2 cmul(float2 a, float2 b) {
  return make_float2(a.x * b.x - a.y * b.y, a.x * b.y + a.y * b.x);
}
__device__ __forceinline__ float2 cadd(float2 a, float2 b) {
  return make_float2(a.x + b.x, a.y + b.y);
}

// ---- CDNA5 async LDS<->global helpers (per-lane addresses, ASYNCcnt-tracked) ----
__device__ __forceinline__ void async_load_b64(const void* gaddr, void* lds) {
#if defined(__HIP_DEVICE_COMPILE__)
  unsigned la = (unsigned)(uintptr_t)lds;
  unsigned long long ga = (unsigned long long)(uintptr_t)gaddr;
  asm volatile("global_load_async_to_lds_b64 %0, %1, off" :: "v"(la), "v"(ga) : "memory");
#else
  *(float2*)lds = *(const float2*)gaddr;
#endif
}
__device__ __forceinline__ void async_store_b128(void* gaddr, const void* lds) {
#if defined(__HIP_DEVICE_COMPILE__)
  unsigned la = (unsigned)(uintptr_t)lds;
  unsigned long long ga = (unsigned long long)(uintptr_t)gaddr;
  asm volatile("global_store_async_from_lds_b128 %0, %1, off" :: "v"(ga), "v"(la) : "memory");
#else
  ((float2*)gaddr)[0] = ((const float2*)lds)[0];
  ((float2*)gaddr)[1] = ((const float2*)lds)[1];
#endif
}
__device__ __forceinline__ void wait_async0() {
#if defined(__HIP_DEVICE_COMPILE__)
  asm volatile("s_wait_asynccnt 0x0" ::: "memory");
#endif
}

// PennyLane Rot(phi, theta, omega) = RZ(omega) RY(theta) RZ(phi)
__device__ __forceinline__ void rotgen(float phi, float th, float om, float2* m) {
  float ct = cosf(0.5f * th), st = sinf(0.5f * th);
  float a = 0.5f * (phi + om), b = 0.5f * (phi - om);
  float ca = cosf(a), sa = sinf(a), cb = cosf(b), sb = sinf(b);
  m[0] = make_float2(ct * ca, -ct * sa);   // e^{-i(phi+om)/2} cos
  m[1] = make_float2(-st * cb, -st * sb);  // -e^{ i(phi-om)/2} sin
  m[2] = make_float2(st * cb, -st * sb);   //  e^{-i(phi-om)/2} sin
  m[3] = make_float2(ct * ca, ct * sa);    //  e^{ i(phi+om)/2} cos
}

// ---------------- prep: v-vectors (embedding+layer0 Rot) and fused 16x16 unitaries -------------
__global__ void prep_kernel(const float* x, const float* params, float2* vtab, float* utab) {
  __shared__ float2 rotm[3][20][4];
  int s = blockIdx.x, tid = threadIdx.x;
  if (tid < 60) {
    int dl = tid / 20, w = tid % 20, d = dl + 1;
    float2 m[4];
    rotgen(params[(d * NQ + w) * 3 + 0], params[(d * NQ + w) * 3 + 1],
           params[(d * NQ + w) * 3 + 2], m);
    for (int k = 0; k < 4; ++k) rotm[dl][w][k] = m[k];
  }
  if (tid < NQ) {
    int w = tid;
    float xv = x[s * NQ + w];
    float c = cosf(0.5f * xv), sn = sinf(0.5f * xv);
    float2 m[4];
    rotgen(params[(0 * NQ + w) * 3 + 0], params[(0 * NQ + w) * 3 + 1],
           params[(0 * NQ + w) * 3 + 2], m);
    float2 r0 = make_float2(c, 0.f), r1 = make_float2(0.f, -sn);  // RX(x)|0>
    vtab[(s * NQ + w) * 2 + 0] = cadd(cmul(m[0], r0), cmul(m[1], r1));
    vtab[(s * NQ + w) * 2 + 1] = cadd(cmul(m[2], r0), cmul(m[3], r1));
  }
  __syncthreads();
  // 6 groups of 4 wires (slot 0 -> row bit i3 / highest bit position), -1 = identity slot
  const int wt[6][4] = { {6, 7, 8, 9}, {10, 11, 12, 13}, {14, -1, -1, -1},
                         {0, 1, 2, 3}, {4, 5, 15, 16},   {-1, 17, 18, 19} };
  for (int e = tid; e < 18 * 256; e += blockDim.x) {
    int u = e >> 8;            // 0..17
    int dl = u / 6, pg = u % 6;
    int ij = e & 255, i = ij >> 4, j = ij & 15;
    float2 prod = make_float2(1.f, 0.f);
    bool zero = false;
#pragma unroll
    for (int q = 0; q < 4; ++q) {
      int ib = (i >> (3 - q)) & 1, jb = (j >> (3 - q)) & 1;
      int w = wt[pg][q];
      if (w < 0) { if (ib != jb) zero = true; }
      else prod = cmul(prod, rotm[dl][w][ib * 2 + jb]);
    }
    if (zero) prod = make_float2(0.f, 0.f);
    int p = (pg >= 3) ? 1 : 0, g = pg % 3;
    float* dst = utab + (size_t)((((dl * 2 + p) * 3 + g) * 64) + s) * 512;
    dst[ij] = prod.x;        // Ur row-major
    dst[256 + ij] = prod.y;  // Ui row-major
  }
}

// ---------------- init: product state, scatter through C(0,1)..C(4,5) of layer 0 ----------------
__global__ __launch_bounds__(256) void init_kernel(const float2* vtab, float2* state) {
  __shared__ float2 tab[32];
  long long blockBase = (long long)blockIdx.x * 8192;
  int s = (int)(blockBase >> 20);
  const float2* v = vtab + s * 40;  // v[w*2 + bit]
  int tid = threadIdx.x;
  if (tid < 32) {  // wires 7..11 <-> bits 12..8
    float2 p = make_float2(1.f, 0.f);
#pragma unroll
    for (int q = 0; q < 5; ++q) p = cmul(p, v[(7 + q) * 2 + ((tid >> (4 - q)) & 1)]);
    tab[tid] = p;
  }
  __syncthreads();
  unsigned blk20 = (unsigned)(blockBase & 0xFFFFF);
  float2 cf = make_float2(1.f, 0.f);
#pragma unroll
  for (int q = 0; q < 7; ++q)  // wires 0..6 <-> bits 19..13 (block-fixed)
    cf = cmul(cf, v[q * 2 + ((blk20 >> (19 - q)) & 1)]);
#pragma unroll
  for (int q = 0; q < 8; ++q)  // wires 12..19 <-> bits 7..0 (= tid)
    cf = cmul(cf, v[(12 + q) * 2 + ((tid >> (7 - q)) & 1)]);
  // scatter map: suffix-xor on bits 19..14 (CNOT arc of layer 0)
  unsigned hi = (blk20 >> 14) & 63;
  unsigned nh = hi ^ (hi >> 1); nh ^= nh >> 2; nh ^= nh >> 4;
  long long outBase = (blockBase & ~0xFC000LL) | ((long long)nh << 14);
  for (int r = 0; r < 32; ++r)
    state[outBase + (long long)r * 256 + tid] = cmul(cf, tab[r]);
}

// ---------------- WMMA 16x16 complex unitary applied on 4 bit positions of an LDS tile ----------
__device__ __forceinline__ v8f wmma_f32(v2f a, v2f b, v8f c) {
  return __builtin_amdgcn_wmma_f32_16x16x4_f32(false, a, false, b, (short)0, c, false, false);
}
__device__ __forceinline__ int insert4(int x, int q0, int q1, int q2, int q3) {
  x = ((x & ~((1 << q0) - 1)) << 1) | (x & ((1 << q0) - 1));
  x = ((x & ~((1 << q1) - 1)) << 1) | (x & ((1 << q1) - 1));
  x = ((x & ~((1 << q2) - 1)) << 1) | (x & ((1 << q2) - 1));
  x = ((x & ~((1 << q3) - 1)) << 1) | (x & ((1 << q3) - 1));
  return x;
}
__device__ __forceinline__ int rowoff(int k, int q0, int q1, int q2, int q3) {
  return ((k & 1) << q0) | (((k >> 1) & 1) << q1) | (((k >> 2) & 1) << q2) | (((k >> 3) & 1) << q3);
}

template <int NB>
__device__ void apply_group(float2* tile, const float* U,
                            int q0, int q1, int q2, int q3, int lane, int wv, int nw) {
  const float* Ur = U;
  const float* Ui = U + 256;
  int n = lane & 15;
  int half = lane >> 4;
  for (int blk = wv; blk < NB; blk += nw) {
    int cb = blk * 16 + n;
    int col = insert4(cb, q0, q1, q2, q3);
    v2f ar[4], ai[4], br[4], bi[4];
#pragma unroll
    for (int kc = 0; kc < 4; ++kc) {
      int kA = kc * 4 + half * 2;  // A/B K-chunk layout: VGPR0 = K0|K2, VGPR1 = K1|K3 by lane half
      int rA = rowoff(kA, q0, q1, q2, q3);
      int rB = rA + (1 << q0);
      float2 s0 = tile[col | rA];
      float2 s1 = tile[col | rB];
      v2f t;
      t.x = s0.x; t.y = s1.x; br[kc] = t;
      t.x = s0.y; t.y = s1.y; bi[kc] = t;
      t.x = Ur[n * 16 + kA]; t.y = Ur[n * 16 + kA + 1]; ar[kc] = t;
      t.x = Ui[n * 16 + kA]; t.y = Ui[n * 16 + kA + 1]; ai[kc] = t;
    }
    v8f rr = {}; v8f ii = {}; v8f ri = {}; v8f ir = {};
#pragma unroll
    for (int kc = 0; kc < 4; ++kc) {
      rr = wmma_f32(ar[kc], br[kc], rr);
      ii = wmma_f32(ai[kc], bi[kc], ii);
      ri = wmma_f32(ar[kc], bi[kc], ri);
      ir = wmma_f32(ai[kc], br[kc], ir);
    }
#pragma unroll
    for (int vv = 0; vv < 8; ++vv) {
      int m = vv + 8 * half;  // C/D layout: VGPR v holds M=v (lanes 0-15) / M=v+8 (lanes 16-31)
      int rm = rowoff(m, q0, q1, q2, q3);
      float2 d;
      d.x = rr[vv] - ii[vv];
      d.y = ri[vv] + ir[vv];
      tile[col | rm] = d;
    }
  }
}

// ---------------- pass A: wires 6..19 local; gather applies C(5,6)..C(18,19) ----------------
#define A_TILE 16384
#define A_SMEM (A_TILE * 8 + 3 * 512 * 4)
__global__ __launch_bounds__(256) void passA_kernel(float2* state, const float* utab, int dl) {
  extern __shared__ char smem[];
  float2* tile = (float2*)smem;
  float* U = (float*)(smem + A_TILE * sizeof(float2));
  int b = blockIdx.x;
  int s = b >> 6, T = b & 63;
  long long gbase = ((long long)s << 20) | ((long long)T << 14);
  int tid = threadIdx.x;
  unsigned w5top = (unsigned)(T & 1) << 13;  // wire-5 control bit (fixed per tile)
  // async gather (CNOT permutation folded into per-lane global addresses)
  for (int c = tid; c < A_TILE; c += blockDim.x) {
    unsigned src = (unsigned)c ^ ((unsigned)c >> 1) ^ w5top;  // inverse suffix-xor map
    async_load_b64(state + gbase + src, tile + c);
  }
  for (int i = tid; i < 3 * 512; i += blockDim.x) {
    int g = i >> 9, r = i & 511;
    U[i] = utab[(size_t)((((dl * 2 + 0) * 3 + g) * 64) + s) * 512 + r];
  }
  wait_async0();
  __syncthreads();
  int lane = tid & 31, wv = tid >> 5, nw = blockDim.x >> 5;
  apply_group<64>(tile, U,          10, 11, 12, 13, lane, wv, nw);  // wires 6..9
  __syncthreads();
  apply_group<64>(tile, U + 512,    6, 7, 8, 9,     lane, wv, nw);  // wires 10..13
  __syncthreads();
  apply_group<64>(tile, U + 1024,   2, 3, 4, 5,     lane, wv, nw);  // wire 14 (+identities)
  __syncthreads();
  // async write-back, contiguous, pair-packed B128 (drains before implicit wait-idle at endpgm)
  for (int c = tid * 2; c < A_TILE; c += 2 * blockDim.x)
    async_store_b128(state + gbase + c, tile + c);
}

// -------- pass B: wires {0..5,15..19} local; gather=C(19,0), scatter=C(0,1)..C(4,5) --------
#define B_TILE 2048
#define B_SMEM (B_TILE * 8 + 3 * 512 * 4)
__global__ __launch_bounds__(256) void passB_kernel(float2* state, const float* utab, int dl) {
  extern __shared__ char smem[];
  float2* tile = (float2*)smem;
  float* U = (float*)(smem + B_TILE * sizeof(float2));
  int b = blockIdx.x;
  int s = b >> 9, t = b & 511;
  long long sbase = (long long)s << 20;
  int tid = threadIdx.x;
  for (int l = tid; l < B_TILE; l += blockDim.x) {
    long long g = ((long long)(l >> 5) << 14) | ((long long)t << 5) | (l & 31);
    g ^= (long long)(l & 1) << 19;  // C(19,0): bit19 ^= bit0
    async_load_b64(state + sbase + g, tile + l);
  }
  for (int i = tid; i < 3 * 512; i += blockDim.x) {
    int g = i >> 9, r = i & 511;
    U[i] = utab[(size_t)((((dl * 2 + 1) * 3 + g) * 64) + s) * 512 + r];
  }
  wait_async0();
  __syncthreads();
  int lane = tid & 31, wv = tid >> 5, nw = blockDim.x >> 5;
  apply_group<8>(tile, U,         7, 8, 9, 10, lane, wv, nw);  // wires 0..3
  __syncthreads();
  apply_group<8>(tile, U + 512,   3, 4, 5, 6,  lane, wv, nw);  // wires 4,5,15,16
  __syncthreads();
  apply_group<8>(tile, U + 1024,  0, 1, 2, 3,  lane, wv, nw);  // wires 17,18,19 (+identity)
  __syncthreads();
  for (int l = tid * 2; l < B_TILE; l += 2 * blockDim.x) {
    unsigned hi = (unsigned)l >> 5;
    unsigned nh = hi ^ (hi >> 1); nh ^= nh >> 2; nh ^= nh >> 4;  // C(0,1)..C(4,5) suffix-xor
    long long g = ((long long)nh << 14) | ((long long)t << 5) | (l & 31);
    async_store_b128(state + sbase + g, tile + l);  // even l: pair stays contiguous & aligned
  }
}

// -------- reduction: trailing CNOTs of layer 3 folded into the sign mask 0x87FFF --------
__global__ __launch_bounds__(256) void reduce_kernel(const float2* state, float* part) {
  int b = blockIdx.x;
  int s = b >> 4, ch = b & 15;
  long long base = ((long long)s << 20) | ((long long)ch << 16);
  float acc = 0.f;
  for (int k = threadIdx.x; k < 65536; k += blockDim.x) {
    unsigned idx = ((unsigned)ch << 16) | (unsigned)k;
    float2 a = state[base + k];
    float v = a.x * a.x + a.y * a.y;
    acc += (__builtin_popcount(idx & 0x87FFFu) & 1) ? -v : v;
  }
  __shared__ float red[256];
  red[threadIdx.x] = acc;
  __syncthreads();
  for (int off = 128; off; off >>= 1) {
    if (threadIdx.x < off) red[threadIdx.x] += red[threadIdx.x + off];
    __syncthreads();
  }
  if (threadIdx.x == 0) part[b] = red[0];
}

__global__ void final_kernel(const float* part, float* out) {
  int s = threadIdx.x;
  if (s < NSAMP) {
    float t = 0.f;
    for (int i = 0; i < 16; ++i) t += part[s * 16 + i];
    out[s] = t;
  }
}

extern "C" void kernel_launch(void* const* d_in, const int* in_sizes, int n_in,
                              void* d_out, int out_size, void* d_ws, size_t ws_size,
                              hipStream_t stream) {
  (void)in_sizes; (void)n_in; (void)out_size; (void)ws_size;
  const float* x = (const float*)d_in[0];
  const float* params = (const float*)d_in[1];
  char* ws = (char*)d_ws;
  float2* state = (float2*)ws;
  const size_t STATE_BYTES = (size_t)NSAMP << 23;  // 64 * 2^20 * 8 = 512 MB
  float2* vtab = (float2*)(ws + STATE_BYTES);
  float* utab = (float*)(ws + STATE_BYTES + (size_t)NSAMP * NQ * 2 * sizeof(float2));
  const size_t UT_BYTES = (size_t)18 * 64 * 512 * 4;
  float* part = (float*)((char*)utab + UT_BYTES);
  float* out = (float*)d_out;

  hipFuncSetAttribute(reinterpret_cast<const void*>(&passA_kernel),
                      hipFuncAttributeMaxDynamicSharedMemorySize, A_SMEM);
  hipFuncSetAttribute(reinterpret_cast<const void*>(&passB_kernel),
                      hipFuncAttributeMaxDynamicSharedMemorySize, B_SMEM);

  prep_kernel<<<NSAMP, 256, 0, stream>>>(x, params, vtab, utab);
  init_kernel<<<8192, 256, 0, stream>>>(vtab, state);
  for (int dl = 0; dl < 3; ++dl) {
    passA_kernel<<<NSAMP * 64, 256, A_SMEM, stream>>>(state, utab, dl);
    passB_kernel<<<NSAMP * 512, 256, B_SMEM, stream>>>(state, utab, dl);
  }
  reduce_kernel<<<NSAMP * 16, 256, 0, stream>>>(state, part);
  final_kernel<<<1, 64, 0, stream>>>(part, out);
}